// GATEncoder_17136919511179
// MI455X (gfx1250) — compile-verified
//
#include <hip/hip_runtime.h>
#include <hip/hip_bf16.h>

// ---------------- problem constants (from reference) ----------------
#define NN   50000      // nodes
#define NE   1600000    // edges
#define INF_ 128        // input feature dim
#define HID  64         // per-head dim
#define HEADS 2
#define HD   (HEADS*HID)   // 128 = fused head*dim
#define NEG_SLOPE 0.2f

typedef __attribute__((ext_vector_type(2))) float v2f;
typedef __attribute__((ext_vector_type(8))) float v8f;

// ---------------- helpers ----------------
__device__ __forceinline__ void atomicMaxF(float* addr, float val) {
    // classic sign-split trick: works for mixed signs when addr is initialized to -inf
    if (val >= 0.0f) atomicMax((int*)addr, __float_as_int(val));
    else             atomicMin((unsigned int*)addr, (unsigned int)__float_as_int(val));
}

// ---------------- kernels ----------------

// WT[n*K + k] = W[k*Nout + n]   (W is [K, Nout] row-major, Nout == 128 always)
__global__ void gat_w_transpose(const float* __restrict__ W, float* __restrict__ WT,
                                int K, int total) {
    int i = blockIdx.x * blockDim.x + threadIdx.x;
    if (i >= total) return;
    int k = i / HD;
    int n = i - k * HD;
    WT[(long)n * K + k] = W[i];
}

// D[M x 128] = A[M x K] * B[K x 128], B given transposed as BT[128 x K].
// One wave per 16x16 tile; 8 waves per block cover the 128 output columns;
// gridDim.x = M/16 (M = 50000 = 3125*16, exact). Uses V_WMMA_F32_16X16X4_F32.
__global__ void __launch_bounds__(256) gat_gemm_wmma_f32(
        const float* __restrict__ A, const float* __restrict__ BT,
        float* __restrict__ D, int K) {
    const int lane = threadIdx.x & 31;
    const int wave = threadIdx.x >> 5;       // 0..7 -> output-column tile
    const int half = lane >> 4;              // 0 or 1
    const int l15  = lane & 15;

    const float* __restrict__ arow = A  + ((long)blockIdx.x * 16 + l15) * K;
    const float* __restrict__ brow = BT + ((long)wave       * 16 + l15) * K;

    v8f c = {};
    for (int kb = 0; kb < K; kb += 4) {
        const int k0 = kb + half * 2;
        v2f a, b;
        a.x = arow[k0]; a.y = arow[k0 + 1];      // A 16x4 f32 fragment (2 VGPRs/lane)
        b.x = brow[k0]; b.y = brow[k0 + 1];      // B 4x16 f32 fragment (2 VGPRs/lane)
        // 8 args: (neg_a, A, neg_b, B, c_mod, C, reuse_a, reuse_b)
        c = __builtin_amdgcn_wmma_f32_16x16x4_f32(false, a, false, b,
                                                  (short)0, c, false, false);
    }
    // C/D layout: lanes 0-15 hold rows v (v=0..7), lanes 16-31 hold rows v+8
    float* __restrict__ drow =
        D + ((long)blockIdx.x * 16 + half * 8) * HD + wave * 16 + l15;
#pragma unroll
    for (int v = 0; v < 8; ++v) drow[(long)v * HD] = c[v];
}

// el[n,h] = feat[n,h,:].al[h,:] ; er likewise.  one thread per (n,h)
__global__ void gat_eler(const float* __restrict__ feat,
                         const float* __restrict__ al, const float* __restrict__ ar,
                         float* __restrict__ el, float* __restrict__ er, int NH) {
    int i = blockIdx.x * blockDim.x + threadIdx.x;
    if (i >= NH) return;
    const int h = i & 1;
    const float* __restrict__ f = feat + (long)(i >> 1) * HD + h * HID;
    const float* __restrict__ la = al + h * HID;
    const float* __restrict__ ra = ar + h * HID;
    float sl = 0.f, sr = 0.f;
#pragma unroll 4
    for (int d = 0; d < HID; ++d) { float v = f[d]; sl += v * la[d]; sr += v * ra[d]; }
    el[i] = sl; er[i] = sr;
}

// init m=-inf, s=0 (NH elems) and agg=0 (NAGG elems)
__global__ void gat_init(float* __restrict__ m, float* __restrict__ s,
                         float* __restrict__ agg, int NH, int NAGG) {
    int i = blockIdx.x * blockDim.x + threadIdx.x;
    if (i < NH) { m[i] = -INFINITY; s[i] = 0.f; }
    if (i < NAGG) agg[i] = 0.f;
}

// e = leaky_relu(el[src]+er[dst]); store to ebuf; atomic segment-max into m[dst]
__global__ void gat_edge_max(const int* __restrict__ src, const int* __restrict__ dst,
                             const float* __restrict__ el, const float* __restrict__ er,
                             float* __restrict__ ebuf, float* __restrict__ m, int E) {
    int e = blockIdx.x * blockDim.x + threadIdx.x;
    if (e >= E) return;
    const int s0 = src[e], d0 = dst[e];
#pragma unroll
    for (int h = 0; h < HEADS; ++h) {
        float v = el[s0 * HEADS + h] + er[d0 * HEADS + h];
        v = v > 0.f ? v : NEG_SLOPE * v;
        ebuf[(long)e * HEADS + h] = v;
        atomicMaxF(&m[d0 * HEADS + h], v);
    }
}

// ex = exp(e - m[dst]); store back; atomic segment-sum into s[dst]
__global__ void gat_edge_exp(const int* __restrict__ dst, const float* __restrict__ m,
                             float* __restrict__ ebuf, float* __restrict__ ssum, int E) {
    int e = blockIdx.x * blockDim.x + threadIdx.x;
    if (e >= E) return;
    const int d0 = dst[e];
#pragma unroll
    for (int h = 0; h < HEADS; ++h) {
        float ex = __expf(ebuf[(long)e * HEADS + h] - m[d0 * HEADS + h]);
        ebuf[(long)e * HEADS + h] = ex;
        unsafeAtomicAdd(&ssum[d0 * HEADS + h], ex);
    }
}

// one wave32 per edge; lane owns 4 of the 128 channels.
// agg[dst] += feat[src] * (ex / s[dst])   (per head)
__global__ void __launch_bounds__(256) gat_edge_scatter(
        const int* __restrict__ src, const int* __restrict__ dst,
        const float* __restrict__ feat, const float* __restrict__ ebuf,
        const float* __restrict__ ssum, float* __restrict__ agg, int E) {
    const int e = blockIdx.x * 8 + (threadIdx.x >> 5);
    if (e >= E) return;
    const int lane = threadIdx.x & 31;
    const int s0 = src[e], d0 = dst[e];
    const int h  = lane >> 4;                 // lanes 0-15 -> head 0, 16-31 -> head 1
    const int c0 = lane * 4;                  // 4 contiguous channels per lane
    const float alpha = ebuf[(long)e * HEADS + h] / ssum[d0 * HEADS + h];

    // prefetch the gather row a few edges ahead (global_prefetch_b8)
    const int e2 = e + 32;
    if (e2 < E) __builtin_prefetch(feat + (long)src[e2] * HD + c0, 0, 0);

    const float4 f = *(const float4*)(feat + (long)s0 * HD + c0);
    float* __restrict__ o = agg + (long)d0 * HD + c0;
    unsafeAtomicAdd(o + 0, f.x * alpha);
    unsafeAtomicAdd(o + 1, f.y * alpha);
    unsafeAtomicAdd(o + 2, f.z * alpha);
    unsafeAtomicAdd(o + 3, f.w * alpha);
}

// out[n,d] = elu( 0.5 * ((agg[n,0,d]+b[0,d]) + (agg[n,1,d]+b[1,d])) )
__global__ void gat_finalize(const float* __restrict__ agg, const float* __restrict__ bias,
                             float* __restrict__ out, int total) {
    int i = blockIdx.x * blockDim.x + threadIdx.x;
    if (i >= total) return;
    const int n = i >> 6, d = i & 63;
    const float v0 = agg[(long)n * HD + d]        + bias[d];
    const float v1 = agg[(long)n * HD + HID + d]  + bias[HID + d];
    const float v  = 0.5f * (v0 + v1);
    out[i] = v > 0.f ? v : (__expf(v) - 1.f);
}

// ---------------- host ----------------
extern "C" void kernel_launch(void* const* d_in, const int* in_sizes, int n_in,
                              void* d_out, int out_size, void* d_ws, size_t ws_size,
                              hipStream_t stream) {
    const float* x   = (const float*)d_in[0];
    const float* W1  = (const float*)d_in[1];
    const float* al1 = (const float*)d_in[2];
    const float* ar1 = (const float*)d_in[3];
    const float* b1  = (const float*)d_in[4];
    const float* W2  = (const float*)d_in[5];
    const float* al2 = (const float*)d_in[6];
    const float* ar2 = (const float*)d_in[7];
    const float* b2  = (const float*)d_in[8];
    const int*   src = (const int*)d_in[9];
    const int*   dst = (const int*)d_in[10];
    float* out = (float*)d_out;

    // bump allocator over workspace (~78 MB used)
    char* wsb = (char*)d_ws;
    size_t off = 0;
    auto alloc = [&](size_t bytes) -> void* {
        void* p = wsb + off;
        off += bytes;
        off = (off + 255) & ~(size_t)255;
        return p;
    };
    float* WT1  = (float*)alloc((size_t)INF_ * HD * 4);        // 64 KB
    float* WT2  = (float*)alloc((size_t)HID  * HD * 4);        // 32 KB
    float* feat = (float*)alloc((size_t)NN * HD * 4);          // 25.6 MB
    float* el   = (float*)alloc((size_t)NN * HEADS * 4);
    float* er   = (float*)alloc((size_t)NN * HEADS * 4);
    float* m    = (float*)alloc((size_t)NN * HEADS * 4);
    float* s    = (float*)alloc((size_t)NN * HEADS * 4);
    float* ebuf = (float*)alloc((size_t)NE * HEADS * 4);       // 12.8 MB
    float* agg  = (float*)alloc((size_t)NN * HD * 4);          // 25.6 MB
    float* h    = (float*)alloc((size_t)NN * HID * 4);         // 12.8 MB

    const int NH   = NN * HEADS;        // 100000
    const int NAGG = NN * HD;           // 6.4M
    const int NOUT = NN * HID;          // 3.2M
    const dim3 B256(256);

    // ---------------- layer 1 ----------------
    gat_w_transpose<<<dim3((INF_ * HD + 255) / 256), B256, 0, stream>>>(W1, WT1, INF_, INF_ * HD);
    gat_gemm_wmma_f32<<<dim3(NN / 16), B256, 0, stream>>>(x, WT1, feat, INF_);
    gat_eler<<<dim3((NH + 255) / 256), B256, 0, stream>>>(feat, al1, ar1, el, er, NH);
    gat_init<<<dim3((NAGG + 255) / 256), B256, 0, stream>>>(m, s, agg, NH, NAGG);
    gat_edge_max<<<dim3((NE + 255) / 256), B256, 0, stream>>>(src, dst, el, er, ebuf, m, NE);
    gat_edge_exp<<<dim3((NE + 255) / 256), B256, 0, stream>>>(dst, m, ebuf, s, NE);
    gat_edge_scatter<<<dim3((NE + 7) / 8), B256, 0, stream>>>(src, dst, feat, ebuf, s, agg, NE);
    gat_finalize<<<dim3((NOUT + 255) / 256), B256, 0, stream>>>(agg, b1, h, NOUT);

    // ---------------- layer 2 ----------------
    gat_w_transpose<<<dim3((HID * HD + 255) / 256), B256, 0, stream>>>(W2, WT2, HID, HID * HD);
    gat_gemm_wmma_f32<<<dim3(NN / 16), B256, 0, stream>>>(h, WT2, feat, HID);
    gat_eler<<<dim3((NH + 255) / 256), B256, 0, stream>>>(feat, al2, ar2, el, er, NH);
    gat_init<<<dim3((NAGG + 255) / 256), B256, 0, stream>>>(m, s, agg, NH, NAGG);
    gat_edge_max<<<dim3((NE + 255) / 256), B256, 0, stream>>>(src, dst, el, er, ebuf, m, NE);
    gat_edge_exp<<<dim3((NE + 255) / 256), B256, 0, stream>>>(dst, m, ebuf, s, NE);
    gat_edge_scatter<<<dim3((NE + 7) / 8), B256, 0, stream>>>(src, dst, feat, ebuf, s, agg, NE);
    gat_finalize<<<dim3((NOUT + 255) / 256), B256, 0, stream>>>(agg, b2, out, NOUT);
}